// LogicTreeConv2d_78537771975314
// MI455X (gfx1250) — compile-verified
//
#include <hip/hip_runtime.h>
#include <math.h>

// ---------------------------------------------------------------------------
// LogicTreeConv2d for MI455X (gfx1250, wave32)
//
// Kernel 1 (coef):  coef = softmax(logits) @ GATE_COEF  via chained
//   V_WMMA_F32_16X16X4_F32 (exact f32, 4 K-chunks of 4).  GATE_COEF values
//   come from a nibble-packed immediate table (branch-free):
//   GT[k] packs (c_n + 2) in bits [4n+3:4n];  value = ((GT>>4n)&15) - 2.
//
// Kernel 2 (gate):  memory-bound gather + gate tree, fully branchless:
//   per-leaf row setup (clamp/predicate/base) hoisted and shared by two
//   4-pixel quads per thread; all loads unconditional from clamped (valid)
//   addresses; zero halo applied with v_cndmask.  grid=(OC,B) keeps the
//   256KB x[b] slab L2-hot; y written once via non-temporal b128 stores.
//   HBM floor: x(16MB) + y(64MB) @ 23.3 TB/s ~= 3.5 us.
// ---------------------------------------------------------------------------

typedef __attribute__((ext_vector_type(2))) float v2f;
typedef __attribute__((ext_vector_type(4))) float v4f;
typedef __attribute__((ext_vector_type(8))) float v8f;

// GT[k]: GATE_COEF row k = (c0,c1,c2,c3), nibble n holds (c_n + 2).
//   c0=b3, c1=b1-b3, c2=b2-b3, c3=b0-b1-b2+b3 with b_i = (k>>i)&1.
__device__ constexpr int GT[16] = {
  0x2222, 0x3222, 0x1232, 0x2232,
  0x1322, 0x2322, 0x0332, 0x1332,
  0x3113, 0x4113, 0x2123, 0x3123,
  0x2213, 0x3213, 0x1223, 0x2223
};

// Branch-free B-fragment element: GATE_COEF[k][n] for n<4, else 0.
__device__ __forceinline__ float bfrag(int T, int n)
{
  const int sh = (n & 7) * 4;              // defined shift (<32)
  const int v  = ((T >> sh) & 15) - 2;     // lshr + and + add
  return (n < 4) ? (float)v : 0.0f;        // cvt + cndmask
}

// One wave per block; 16 (oc,node) rows each:
//   D(16x16) = sum_{kc} A_kc(16x4) * B_kc(4x16),  B = GATE_COEF zero-padded.
__global__ __launch_bounds__(32) void coef_kernel(const float* __restrict__ logits,
                                                  float* __restrict__ coef)
{
  __shared__ float P[16][18];          // probs tile, padded stride
  const int lane    = threadIdx.x;     // 0..31
  const int rowBase = blockIdx.x * 16; // row = oc*7 + node

  // --- softmax over 16 logits, one row per lane (lanes 0..15) -------------
  if (lane < 16) {
    const v4f* Lp = (const v4f*)(logits + (size_t)(rowBase + lane) * 16);
    float v[16];
#pragma unroll
    for (int i = 0; i < 4; ++i) {
      const v4f q = Lp[i];
      v[4 * i + 0] = q.x; v[4 * i + 1] = q.y;
      v[4 * i + 2] = q.z; v[4 * i + 3] = q.w;
    }
    float mx = -INFINITY;
#pragma unroll
    for (int i = 0; i < 16; ++i) mx = fmaxf(mx, v[i]);
    float s = 0.f;
#pragma unroll
    for (int i = 0; i < 16; ++i) { v[i] = expf(v[i] - mx); s += v[i]; }
    const float inv = 1.0f / s;
#pragma unroll
    for (int i = 0; i < 16; ++i) P[lane][i] = v[i] * inv;
  }
  __syncthreads();

  // --- chained f32 WMMA, 4 chunks of K=4 ----------------------------------
  // A (16x4 f32): lanes 0-15: M=lane, VGPR0/1 = K=0/1; lanes 16-31: K=2/3.
  // B (4x16 f32): VGPR0 = row K=0 (lanes 0-15) / K=2 (lanes 16-31), VGPR1 = K=1/3.
  const int n  = lane & 15;   // A row M / B-D column N
  const int hi = lane >> 4;

  v8f d = {0.f, 0.f, 0.f, 0.f, 0.f, 0.f, 0.f, 0.f};
#pragma unroll
  for (int kc = 0; kc < 4; ++kc) {
    const int k0 = kc * 4 + hi * 2;
    v2f a;
    a.x = P[n][k0];
    a.y = P[n][k0 + 1];
    // GT index constant after unroll: one cndmask between two literals.
    const int Tx = hi ? GT[kc * 4 + 2] : GT[kc * 4 + 0];
    const int Ty = hi ? GT[kc * 4 + 3] : GT[kc * 4 + 1];
    v2f b;
    b.x = bfrag(Tx, n);
    b.y = bfrag(Ty, n);
    d = __builtin_amdgcn_wmma_f32_16x16x4_f32(false, a, false, b,
                                              (short)0, d, false, false);
  }

  // --- store D columns 0..3: VGPR v holds M = v + 8*hi, N = lane&15 -------
  if (n < 4) {
#pragma unroll
    for (int v = 0; v < 8; ++v)
      coef[(size_t)(rowBase + v + 8 * hi) * 4 + n] = d[v];
  }
}

__device__ __forceinline__ float gate(const float c[4], float a, float b)
{
  // c0 + c1*a + c2*b + c3*a*b
  return fmaf(fmaf(c[3], a, c[2]), b, fmaf(c[1], a, c[0]));
}

// grid = (OC, B), block = 128 threads, 8 consecutive pixels (2 quads) each.
__global__ __launch_bounds__(128) void gate_kernel(const float* __restrict__ x,
                                                   const int*   __restrict__ leaf_idx,
                                                   const float* __restrict__ coef,
                                                   float*       __restrict__ y)
{
  const int oc = blockIdx.x;   // 0..255
  const int b  = blockIdx.y;   // 0..63
  const int t  = threadIdx.x;  // 0..127

  // Uniform per block: decode the 8 leaf indices into (channel, dy, dx).
  int ch[8], dy[8], dx[8];
#pragma unroll
  for (int l = 0; l < 8; ++l) {
    const int ii = leaf_idx[oc * 8 + l];   // 0..575
    ch[l] = ii / 9;
    const int k  = ii - ch[l] * 9;
    const int ky = k / 3;
    dy[l] = ky - 1;
    dx[l] = (k - ky * 3) - 1;
  }

  // Uniform per block: 7x4 gate coefficients.
  float cf[7][4];
  const float* cfp = coef + (size_t)oc * 28;
#pragma unroll
  for (int nd = 0; nd < 7; ++nd)
#pragma unroll
    for (int j = 0; j < 4; ++j)
      cf[nd][j] = cfp[nd * 4 + j];

  const int p0   = t * 8;      // pixel base, 0..1016 (8 px in one image row)
  const int row  = p0 >> 5;    // 0..31
  const int col0 = p0 & 31;    // 0,8,16,24
  const float* xb = x + (size_t)b * 64 * 1024;

  // Per-leaf row setup, shared by both quads: clamp row, predicate, base ptr.
  const float* srow[8];
  bool rowok[8];
#pragma unroll
  for (int l = 0; l < 8; ++l) {
    const int yy  = row + dy[l];
    const int yyc = min(max(yy, 0), 31);               // v_med3_i32
    rowok[l] = ((unsigned)yy < 32u);
    srow[l]  = xb + (size_t)ch[l] * 1024 + yyc * 32;
  }

  float* ydst = y + ((size_t)(b * 256 + oc) * 1024 + p0);

#pragma unroll
  for (int q = 0; q < 2; ++q) {
    const int col = col0 + q * 4;                      // 0,4,...,28

    // Branchless gather: for col multiple of 4, xf+1 and xf+2 are always in
    // [0,31]; only xf (dx=-1) and xf+3 (dx=+1) can be OOB -> clamp + mask.
    float lv[8][4];
#pragma unroll
    for (int l = 0; l < 8; ++l) {
      const int xf = col + dx[l];                      // in [-1, 29]
      const int x0 = max(xf, 0);
      const int x3 = min(xf + 3, 31);
      const float* s = srow[l];

      const float v0 = s[x0];                          // unconditional loads
      const float v1 = s[xf + 1];
      const float v2 = s[xf + 2];
      const float v3 = s[x3];

      const bool ok = rowok[l];
      lv[l][0] = (ok && xf >= 0)      ? v0 : 0.0f;     // v_cndmask
      lv[l][1] = ok                   ? v1 : 0.0f;
      lv[l][2] = ok                   ? v2 : 0.0f;
      lv[l][3] = (ok && x3 == xf + 3) ? v3 : 0.0f;
    }

    // Depth-3 gate tree: nodes 0-3 (level 0), 4-5 (level 1), 6 (root).
    float r[4];
#pragma unroll
    for (int j = 0; j < 4; ++j) {
      const float n0 = gate(cf[0], lv[0][j], lv[1][j]);
      const float n1 = gate(cf[1], lv[2][j], lv[3][j]);
      const float n2 = gate(cf[2], lv[4][j], lv[5][j]);
      const float n3 = gate(cf[3], lv[6][j], lv[7][j]);
      const float m0 = gate(cf[4], n0, n1);
      const float m1 = gate(cf[5], n2, n3);
      r[j] = gate(cf[6], m0, m1);
    }

    // Streaming (non-temporal) aligned 16B store: y is write-once.
    v4f out;
    out.x = r[0]; out.y = r[1]; out.z = r[2]; out.w = r[3];
    __builtin_nontemporal_store(out, (v4f*)(ydst + q * 4));
  }
}

extern "C" void kernel_launch(void* const* d_in, const int* in_sizes, int n_in,
                              void* d_out, int out_size, void* d_ws, size_t ws_size,
                              hipStream_t stream)
{
  (void)in_sizes; (void)n_in; (void)out_size; (void)ws_size;
  const float* x      = (const float*)d_in[0];  // (64,64,32,32) f32
  const float* logits = (const float*)d_in[1];  // (256,7,16)    f32
  const int*   leaves = (const int*)  d_in[2];  // (256,8)       int
  float*       y      = (float*)d_out;          // (64,256,32,32) f32
  float*       coef   = (float*)d_ws;           // (256,7,4) f32 scratch = 28 KB

  // 1792 coef rows -> 112 single-wave blocks (WMMA f32 16x16x4 path).
  coef_kernel<<<112, 32, 0, stream>>>(logits, coef);

  // Main streaming kernel: oc fastest in grid.x so x[b] stays L2-hot.
  dim3 grid(256, 64);
  gate_kernel<<<grid, 128, 0, stream>>>(x, leaves, coef, y);
}